// LIFSpike_79826262163576
// MI455X (gfx1250) — compile-verified
//
#include <hip/hip_runtime.h>
#include <cmath>

// LIF spike scan: out[t] = ((TAU*mem*(1-prev_spike) + x[t]) > tanh(w[c]))
// Shapes: x [T=16, B=32, C=64, H=32, W=32] f32, w [64] f32, out same as x.
//
// Roofline: 128 MiB read + 128 MiB write @ 23.3 TB/s => ~11.5 us floor;
// ~4 VALU/element is far below the VALU roof -> purely bandwidth-bound.
// Strategy:
//   * prologue kernel: thre[c] = tanh(w[c])  (64 threads, once) -> d_ws
//   * main kernel: 1 thread per float4 site, all 16 global_load_b128 issued
//     up-front (deep MLP), reset-applied-membrane recurrence = 4 VALU/elem
//     (fma + cmp + 2x cndmask sharing one VCC), wave-uniform threshold in
//     an SGPR via readfirstlane, non-temporal b128 stores so the replayed
//     input stream stays resident in the 192 MB L2.

typedef float v4f __attribute__((ext_vector_type(4)));

#define TAU_F     0.25f
#define T_STEPS   16
#define C_CH      64
#define N_SPAT    (32 * 64 * 32 * 32)    // B*C*H*W = 2,097,152 floats
#define BLOCK_SZ  256

// ---- prologue: 64 thresholds, computed once per launch into workspace ----
__global__ void lif_thre_kernel(const float* __restrict__ w,
                                float* __restrict__ thre)
{
    const int c = threadIdx.x;           // launched with exactly C_CH threads
    thre[c] = tanhf(w[c]);
}

// ---- main streaming scan ----
__global__ __launch_bounds__(BLOCK_SZ) void
lif_scan_kernel(const float* __restrict__ x,
                const float* __restrict__ thre_tab,
                float* __restrict__ out)
{
    const int    tid  = blockIdx.x * BLOCK_SZ + threadIdx.x;
    const size_t base = (size_t)tid * 4;              // flat spatial float index

    // Issue the 16 independent streaming loads first: 16 global_load_b128
    // in flight per lane before any dependent compute.
    v4f xv[T_STEPS];
#pragma unroll
    for (int t = 0; t < T_STEPS; ++t) {
        xv[t] = *(const v4f*)(x + (size_t)t * N_SPAT + base);
    }

    // Channel of this float4 (H*W = 1024, multiple of 4 -> no channel split).
    const int c = (int)((base >> 10) & (C_CH - 1));
    float thre  = thre_tab[c];           // 256 B table -> L1 hit
    // A wave32 spans 128 consecutive floats; channel blocks are 1024 floats
    // and 128 | 1024, so thre is wave-uniform -> pin it in an SGPR.
    thre = __int_as_float(__builtin_amdgcn_readfirstlane(__float_as_int(thre)));

    // m = reset-applied membrane from previous step: mem*(1-prev_spike).
    v4f m = (v4f)0.0f;

#pragma unroll
    for (int t = 0; t < T_STEPS; ++t) {
        v4f u = TAU_F * m + xv[t];       // v_fma per component

        const bool gx = u.x > thre;      // one v_cmp feeds both cndmasks
        const bool gy = u.y > thre;
        const bool gz = u.z > thre;
        const bool gw = u.w > thre;

        v4f o;
        o.x = gx ? 1.0f : 0.0f;
        o.y = gy ? 1.0f : 0.0f;
        o.z = gz ? 1.0f : 0.0f;
        o.w = gw ? 1.0f : 0.0f;

        m.x = gx ? 0.0f : u.x;           // reset-by-spike, applied eagerly
        m.y = gy ? 0.0f : u.y;
        m.z = gz ? 0.0f : u.z;
        m.w = gw ? 0.0f : u.w;

        // Non-temporal store: keep the 128 MiB output stream from evicting
        // the 128 MiB input out of the 192 MB L2 (graph-replay friendly).
        __builtin_nontemporal_store(o, (v4f*)(out + (size_t)t * N_SPAT + base));
    }
}

extern "C" void kernel_launch(void* const* d_in, const int* in_sizes, int n_in,
                              void* d_out, int out_size, void* d_ws, size_t ws_size,
                              hipStream_t stream)
{
    (void)in_sizes; (void)n_in; (void)ws_size; (void)out_size;

    const float* x    = (const float*)d_in[0];
    const float* w    = (const float*)d_in[1];
    float*       out  = (float*)d_out;
    float*       thre = (float*)d_ws;               // 64 floats of scratch

    lif_thre_kernel<<<1, C_CH, 0, stream>>>(w, thre);

    const int n_vec4 = N_SPAT / 4;                  // 524,288 float4 sites
    const int blocks = n_vec4 / BLOCK_SZ;           // 2048 blocks of 8 wave32s
    lif_scan_kernel<<<blocks, BLOCK_SZ, 0, stream>>>(x, thre, out);
}